// CCSDS123Compressor_2216203124849
// MI455X (gfx1250) — compile-verified
//
#include <hip/hip_runtime.h>
#include <cstdint>
#include <cstddef>

// CCSDS-123 lossless predictor, MI455X (gfx1250).
// Memory-bound: ~235 MB in, ~1.41 GB out => ~70 us at 23.3 TB/s.
// Strategy: async global->LDS row staging (ASYNCcnt), LDS-sourced stencil,
// non-temporal 128-bit stores for the six write-once output planes.

#define DIM_Z 224
#define DIM_Y 512
#define DIM_X 512

typedef float v4f __attribute__((ext_vector_type(4)));

__device__ __forceinline__ void async_load_row_b128(void* lds_ptr, const void* gptr) {
    // GV mode: 64-bit vaddr, LDS destination address in a VGPR (low 32 bits of
    // the generic LDS pointer == wave-relative LDS offset per flat-aperture rules).
    uint32_t lds_off = (uint32_t)(uintptr_t)lds_ptr;
    uint64_t gaddr   = (uint64_t)(uintptr_t)gptr;
    asm volatile("global_load_async_to_lds_b128 %0, %1, off"
                 :
                 : "v"(lds_off), "v"(gaddr)
                 : "memory");
}

__device__ __forceinline__ void wait_async_all() {
    asm volatile("s_wait_asynccnt 0" ::: "memory");
}

__global__ __launch_bounds__(128) void ccsds123_predict_kernel(
    const float* __restrict__ img,   // [Z, Y, X]
    float* __restrict__ out)         // 6 planes of [Z, Y, X], concatenated
{
    const int y = blockIdx.x;        // 0..511
    const int z = blockIdx.y;        // 0..223
    const int t = threadIdx.x;       // 0..127, each owns 4 consecutive x

    __shared__ float cur  [DIM_X];   // s(z, y, :)
    __shared__ float north[DIM_X];   // s(z, y-1, :)   (valid iff y > 0)
    __shared__ float prevb[DIM_X];   // s(z-1, y, :)   (valid iff z > 0)

    const size_t planeN  = (size_t)DIM_Z * DIM_Y * DIM_X;
    const size_t rowBase = ((size_t)z * DIM_Y + y) * DIM_X;
    const int    x4      = t * 4;

    // ---- async stage: 16 B per lane per stream (block-uniform predicates) ----
    async_load_row_b128(&cur[x4], img + rowBase + x4);
    if (y > 0) {
        async_load_row_b128(&north[x4], img + rowBase - DIM_X + x4);
    }
    if (z > 0) {
        async_load_row_b128(&prevb[x4], img + rowBase - (size_t)DIM_Y * DIM_X + x4);
    }
    wait_async_all();
    __syncthreads();

    // ---- stencil + predictor (all neighbor traffic from LDS) ----
    float pr[4], rs[4], mp[4], rc[4];
    #pragma unroll
    for (int i = 0; i < 4; ++i) {
        const int   x = x4 + i;
        const float s = cur[x];

        // neighbor-oriented wide local sum with CCSDS edge cases
        float sigma;
        if (y == 0) {
            sigma = (x == 0) ? 0.0f : 4.0f * cur[x - 1];              // top row: 4*W
        } else if (x == 0) {
            sigma = 2.0f * (north[0] + north[1]);                     // left col: 2*(N+NE)
        } else if (x == DIM_X - 1) {
            sigma = cur[x - 1] + north[x - 1] + 2.0f * north[x];      // right col: W+NW+2N
        } else {
            sigma = cur[x - 1] + north[x - 1] + north[x] + north[x + 1];
        }

        const float spatial = sigma * 0.25f;
        float pred = (z == 0) ? spatial : 0.5f * (spatial + prevb[x]);
        if (y == 0 && x == 0) {                                       // origin sample
            pred = (z == 0) ? 0.0f : prevb[0];
        }

        const float res = s - pred;                                   // lossless: q-res == res
        const float qf  = __builtin_rintf(res);                       // RNE == jnp.round
        const int   q   = (int)qf;
        const int   m   = (q >= 0) ? (2 * q) : (-2 * q - 1);          // zigzag map

        float recon = pred + res;
        recon = fminf(fmaxf(recon, -32768.0f), 32767.0f);

        pr[i] = pred;
        rs[i] = res;
        mp[i] = (float)m;
        rc[i] = recon;
    }

    // ---- six write-once planes: 128-bit non-temporal stores ----
    const size_t o = rowBase + x4;
    v4f v;

    v = (v4f){pr[0], pr[1], pr[2], pr[3]};
    __builtin_nontemporal_store(v, (v4f*)(out + 0 * planeN + o));     // predictions

    v = (v4f){rs[0], rs[1], rs[2], rs[3]};
    __builtin_nontemporal_store(v, (v4f*)(out + 1 * planeN + o));     // residuals
    __builtin_nontemporal_store(v, (v4f*)(out + 2 * planeN + o));     // quantized (==residuals)

    v = (v4f){mp[0], mp[1], mp[2], mp[3]};
    __builtin_nontemporal_store(v, (v4f*)(out + 3 * planeN + o));     // mapped indices

    v = (v4f){rc[0], rc[1], rc[2], rc[3]};
    __builtin_nontemporal_store(v, (v4f*)(out + 4 * planeN + o));     // sample reps (==recon)
    __builtin_nontemporal_store(v, (v4f*)(out + 5 * planeN + o));     // reconstructed
}

extern "C" void kernel_launch(void* const* d_in, const int* in_sizes, int n_in,
                              void* d_out, int out_size, void* d_ws, size_t ws_size,
                              hipStream_t stream) {
    (void)in_sizes; (void)n_in; (void)out_size; (void)d_ws; (void)ws_size;
    const float* img = (const float*)d_in[0];
    float*       out = (float*)d_out;

    dim3 grid(DIM_Y, DIM_Z);   // one block per (z, y) row
    dim3 block(128);           // 4 wave32; each lane owns 4 consecutive x
    ccsds123_predict_kernel<<<grid, block, 0, stream>>>(img, out);
}